// EmbeddingToExpression_45157286150943
// MI455X (gfx1250) — compile-verified
//
#include <hip/hip_runtime.h>
#include <stdint.h>

// Problem dims
#define N_CELLS   2000
#define N_REGIONS 5000
#define DDIM      5
// Blocking
#define RB     250                 // regions per block (20 blocks * 250 = 5000 exactly)
#define CB     80                  // cells per block   (25 blocks * 80  = 2000 exactly)
#define CHUNK  4                   // cells per TDM chunk
#define NCHUNK (CB / CHUNK)        // 20
#define TILE_W (RB * DDIM)         // 1250 floats per cell-row in a tile
#define BUF_FLOATS (TILE_W * CHUNK) // 5000 floats = 20000 B per buffer
#define BUF_BYTES  (BUF_FLOATS * 4)
#define ROW_ELTS   (N_REGIONS * DDIM) // 25000 elements per cell row in the full tensor

typedef unsigned int u32x4 __attribute__((ext_vector_type(4)));
typedef int          i32x4 __attribute__((ext_vector_type(4)));
typedef int          i32x8 __attribute__((ext_vector_type(8)));

// Five hardware tanh (TRANS) ops in one block: each v_tanh covers the previous
// one's 1-op TRANS result hazard; only the last needs a trailing v_nop.
// Early-clobber outputs: each tanh writes before later inputs are read.
__device__ __forceinline__ void tanh5(const float* u, float* t) {
  asm volatile(
      "v_tanh_f32 %0, %5\n\t"
      "v_tanh_f32 %1, %6\n\t"
      "v_tanh_f32 %2, %7\n\t"
      "v_tanh_f32 %3, %8\n\t"
      "v_tanh_f32 %4, %9\n\t"
      "v_nop"
      : "=&v"(t[0]), "=&v"(t[1]), "=&v"(t[2]), "=&v"(t[3]), "=&v"(t[4])
      : "v"(u[0]), "v"(u[1]), "v"(u[2]), "v"(u[3]), "v"(u[4]));
}

// Issue one TDM 2D tile load: [CHUNK cells x TILE_W floats], row stride ROW_ELTS.
// D# packing per CDNA5 ISA section 8.3/8.4 (group0: count/lds/global/type, group1: dims).
__device__ __forceinline__ void tdm_load_tile(uint32_t lds_addr, uint64_t gaddr) {
  u32x4 g0;
  g0[0] = 1u;                                        // count=1, is_restore=0, gather=0
  g0[1] = lds_addr;                                  // LDS byte address
  g0[2] = (uint32_t)(gaddr & 0xFFFFFFFFu);           // global_addr[31:0]
  g0[3] = (uint32_t)((gaddr >> 32) & 0x1FFFFFFu)     // global_addr[56:32]
        | (2u << 30);                                // type=2 ("image")
  i32x8 g1;
  g1[0] = (int)(2u << 16);                           // data_size=2 (4 bytes); mask=0; no pad/iter
  g1[1] = (int)((uint32_t)ROW_ELTS << 16);           // tensor_dim0[15:0] in dw1[31:16]
  g1[2] = (int)(((uint32_t)ROW_ELTS >> 16) & 0xFFFFu) // tensor_dim0[31:16]
        | (int)((uint32_t)N_CELLS << 16);            // tensor_dim1[15:0]
  g1[3] = (int)(((uint32_t)N_CELLS >> 16) & 0xFFFFu) // tensor_dim1[31:16]
        | (int)((uint32_t)TILE_W << 16);             // tile_dim0 = 1250
  g1[4] = (int)CHUNK;                                // tile_dim1 = 4, tile_dim2 = 0
  g1[5] = (int)ROW_ELTS;                             // tensor_dim0_stride[31:0] = 25000
  g1[6] = 0;                                         // stride0[47:32]=0, stride1[15:0]=0
  g1[7] = 0;
  i32x4 gz4 = {0, 0, 0, 0};                          // groups 2/3 unused (2D tile)
  i32x8 gz8 = {0, 0, 0, 0, 0, 0, 0, 0};              // trailing group (clang-23 6-arg form)
  __builtin_amdgcn_tensor_load_to_lds(g0, g1, gz4, gz4, gz8, 0);
}

__global__ __launch_bounds__(256)
void etx_kernel(const float* __restrict__ emb, const long long* __restrict__ roi,
                const float* __restrict__ W1, const float* __restrict__ b1,
                const float* __restrict__ W2, const float* __restrict__ b2,
                const float* __restrict__ Wf, float* __restrict__ out) {
  __shared__ float smem[2 * BUF_FLOATS];             // 40 KB double buffer

  const int tid = threadIdx.x;
  const int r0  = blockIdx.x * RB;
  const int c0  = blockIdx.y * CB;
  const bool active = (tid < RB);
  const int r = r0 + tid;

  // Per-lane region weights live in VGPRs for the whole kernel (65 floats).
  float w1[DDIM][DDIM], w2[DDIM][DDIM], b1v[DDIM], b2v[DDIM], wfv[DDIM];
  if (active) {
    const int wr = (int)roi[r];                      // head gather (identity here)
    const float* p1 = W1 + (size_t)wr * 25;
    const float* p2 = W2 + (size_t)wr * 25;
#pragma unroll
    for (int d = 0; d < DDIM; ++d)
#pragma unroll
      for (int e = 0; e < DDIM; ++e) { w1[d][e] = p1[d * 5 + e]; w2[d][e] = p2[d * 5 + e]; }
#pragma unroll
    for (int e = 0; e < DDIM; ++e) {
      b1v[e] = b1[(size_t)wr * 5 + e];
      b2v[e] = b2[(size_t)wr * 5 + e];
      wfv[e] = Wf[(size_t)wr * 5 + e];
    }
  }

  const uint32_t lds_base = (uint32_t)(uintptr_t)(&smem[0]); // LDS offset (flat low 32 bits)
  const uint64_t gbase = (uint64_t)(uintptr_t)emb;

  // Prologue: wave 0 kicks off chunk 0 (TENSOR ops ignore EXEC -> only one wave may issue).
  if (tid < 32) {
    uint64_t ga = gbase + 4ull * ((uint64_t)c0 * ROW_ELTS + (uint64_t)r0 * DDIM);
    tdm_load_tile(lds_base, ga);
  }

  for (int it = 0; it < NCHUNK; ++it) {
    const int cur = it & 1;
    if (tid < 32) __builtin_amdgcn_s_wait_tensorcnt(0); // chunk `it` resident in LDS
    __syncthreads();                                    // publish; everyone done with other buffer
    if ((tid < 32) && (it + 1 < NCHUNK)) {
      uint64_t ga = gbase + 4ull * ((uint64_t)(c0 + (it + 1) * CHUNK) * ROW_ELTS
                                    + (uint64_t)r0 * DDIM);
      tdm_load_tile(lds_base + (uint32_t)((1 - cur) * BUF_BYTES), ga);
    }
    if (active) {
      const float* buf = &smem[cur * BUF_FLOATS];
#pragma unroll
      for (int cc = 0; cc < CHUNK; ++cc) {
        const float* xp = buf + cc * TILE_W + tid * DDIM;
        float x[DDIM];
#pragma unroll
        for (int d = 0; d < DDIM; ++d) x[d] = xp[d];   // LDS, stride-5 words: conflict-free

        // Layer 1: a = W1^T x + b1 ; h = gelu(a) (tanh form, hw v_tanh_f32)
        float a1[DDIM], u1[DDIM], t1[DDIM], h[DDIM];
#pragma unroll
        for (int e = 0; e < DDIM; ++e) {
          float a = b1v[e];
#pragma unroll
          for (int d = 0; d < DDIM; ++d) a = fmaf(x[d], w1[d][e], a);
          a1[e] = a;
          u1[e] = 0.7978845608028654f * a * fmaf(0.044715f * a, a, 1.0f);
        }
        tanh5(u1, t1);
#pragma unroll
        for (int e = 0; e < DDIM; ++e) h[e] = 0.5f * a1[e] * (1.0f + t1[e]);

        // Layer 2
        float a2[DDIM], u2[DDIM], t2[DDIM], g[DDIM];
#pragma unroll
        for (int e = 0; e < DDIM; ++e) {
          float a = b2v[e];
#pragma unroll
          for (int d = 0; d < DDIM; ++d) a = fmaf(h[d], w2[d][e], a);
          a2[e] = a;
          u2[e] = 0.7978845608028654f * a * fmaf(0.044715f * a, a, 1.0f);
        }
        tanh5(u2, t2);
#pragma unroll
        for (int e = 0; e < DDIM; ++e) g[e] = 0.5f * a2[e] * (1.0f + t2[e]);

        // Final per-region head: out = g . Wf[r]
        float acc = 0.0f;
#pragma unroll
        for (int d = 0; d < DDIM; ++d) acc = fmaf(g[d], wfv[d], acc);
        out[(uint64_t)(c0 + it * CHUNK + cc) * N_REGIONS + r] = acc; // coalesced
      }
    }
  }
}

extern "C" void kernel_launch(void* const* d_in, const int* in_sizes, int n_in,
                              void* d_out, int out_size, void* d_ws, size_t ws_size,
                              hipStream_t stream) {
  const float*     emb = (const float*)d_in[0];
  const long long* roi = (const long long*)d_in[1];   // int64 regions_oi
  const float*     W1  = (const float*)d_in[2];
  const float*     b1  = (const float*)d_in[3];
  const float*     W2  = (const float*)d_in[4];
  const float*     b2  = (const float*)d_in[5];
  const float*     Wf  = (const float*)d_in[6];
  float*           out = (float*)d_out;

  dim3 grid(N_REGIONS / RB, N_CELLS / CB);            // 20 x 25 = 500 workgroups
  etx_kernel<<<grid, 256, 0, stream>>>(emb, roi, W1, b1, W2, b2, Wf, out);
}